// ActGINLayer_53060025975246
// MI455X (gfx1250) — compile-verified
//
#include <hip/hip_runtime.h>

typedef __attribute__((ext_vector_type(2))) float v2f;
typedef __attribute__((ext_vector_type(8))) float v8f;

#define N_NODES   50000
#define N_EDGES   800000
#define D         128
#define BATCHSIZE 64
#define ROWS_PER_BLOCK 16
#define LDS_STRIDE 132   // 132 % 64 banks == 4 -> conflict-free strided A reads

// ---------------------------------------------------------------------------
// Kernel 0: zero the agg workspace and the pooled2 region of d_out.
// ---------------------------------------------------------------------------
__global__ void zero_kernel(float* __restrict__ agg, float* __restrict__ pooled2) {
    int i = blockIdx.x * blockDim.x + threadIdx.x;
    const int nagg = N_NODES * D;
    if (i < nagg) {
        agg[i] = 0.0f;
    } else if (i < nagg + BATCHSIZE * D) {
        pooled2[i - nagg] = 0.0f;
    }
}

// ---------------------------------------------------------------------------
// Kernel 1: edge scatter-add.  One wave32 per edge: 32 lanes x float4 = 512B
// coalesced gather of x[src], then global_atomic_add_f32 x4 into agg[dst].
// ---------------------------------------------------------------------------
__global__ __launch_bounds__(256) void edge_scatter(const float* __restrict__ x,
                                                    const int*   __restrict__ edge_index,
                                                    float*       __restrict__ agg) {
    long long t = (long long)blockIdx.x * blockDim.x + threadIdx.x;
    int e    = (int)(t >> 5);
    int lane = (int)(t & 31);
    if (e >= N_EDGES) return;
    int src = edge_index[e];            // row 0 of edge_index
    int dst = edge_index[N_EDGES + e];  // row 1 of edge_index
    const float4 v = *(const float4*)(x + (long long)src * D + lane * 4);
    float* p = agg + (long long)dst * D + lane * 4;
    atomicAdd(p + 0, v.x);
    atomicAdd(p + 1, v.y);
    atomicAdd(p + 2, v.z);
    atomicAdd(p + 3, v.w);
}

// ---------------------------------------------------------------------------
// Kernel 2: fused  out = relu((1+eps)*x + agg)  ->  d_out
//                  pooled = out @ W + b         (V_WMMA_F32_16X16X4_F32)
//                  pooled2[batch[m]] += pooled  (atomic f32)
// 256 threads = 8 waves; 16 rows per block (50000 = 3125 * 16, no tail so
// EXEC is all-ones at every WMMA, as the ISA requires).
// ---------------------------------------------------------------------------
__global__ __launch_bounds__(256) void fused_out_gemm_pool(
        const float* __restrict__ x,
        const float* __restrict__ eps,
        const float* __restrict__ W,      // [D, D] row-major (K rows, N cols)
        const float* __restrict__ bvec,   // [D]
        const int*   __restrict__ batch,  // [N_NODES]
        const float* __restrict__ agg,    // [N_NODES, D]
        float*       __restrict__ out,    // [N_NODES, D]
        float*       __restrict__ pooled2)// [BATCHSIZE, D]
{
    __shared__ float s_out[ROWS_PER_BLOCK * LDS_STRIDE];
    __shared__ int   s_batch[ROWS_PER_BLOCK];

    const int tid  = threadIdx.x;
    const int row0 = blockIdx.x * ROWS_PER_BLOCK;
    const float e1 = 1.0f + eps[0];

    // ---- Stage 1: relu tile -> LDS + stream to global out -----------------
    {
        int r  = tid >> 4;           // 0..15
        int c0 = (tid & 15) * 8;     // 0..120 step 8
        long long g = (long long)(row0 + r) * D;
        float4 xa = *(const float4*)(x   + g + c0);
        float4 xb = *(const float4*)(x   + g + c0 + 4);
        float4 ga = *(const float4*)(agg + g + c0);
        float4 gb = *(const float4*)(agg + g + c0 + 4);
        float4 oa, ob;
        oa.x = fmaxf(e1 * xa.x + ga.x, 0.0f);
        oa.y = fmaxf(e1 * xa.y + ga.y, 0.0f);
        oa.z = fmaxf(e1 * xa.z + ga.z, 0.0f);
        oa.w = fmaxf(e1 * xa.w + ga.w, 0.0f);
        ob.x = fmaxf(e1 * xb.x + gb.x, 0.0f);
        ob.y = fmaxf(e1 * xb.y + gb.y, 0.0f);
        ob.z = fmaxf(e1 * xb.z + gb.z, 0.0f);
        ob.w = fmaxf(e1 * xb.w + gb.w, 0.0f);
        *(float4*)(&s_out[r * LDS_STRIDE + c0])     = oa;
        *(float4*)(&s_out[r * LDS_STRIDE + c0 + 4]) = ob;
        *(float4*)(out + g + c0)     = oa;
        *(float4*)(out + g + c0 + 4) = ob;
        if (tid < ROWS_PER_BLOCK) s_batch[tid] = batch[row0 + tid];
    }
    __syncthreads();

    // ---- Stage 2: 16x128 @ 128x128 GEMM via f32 WMMA ----------------------
    // Wave w owns output columns n0 = 16*w .. +16.
    const int wave  = tid >> 5;
    const int lane  = tid & 31;
    const int n0    = wave * 16;
    const int m     = lane & 15;          // A-fragment row
    const int khalf = (lane >> 4) * 2;    // lanes 0-15: K base +0 ; 16-31: +2

    v8f acc = {};
    #pragma unroll
    for (int k0 = 0; k0 < D; k0 += 4) {
        v2f a, bm;
        // A (16x4): lane holds out[m][k0+khalf], out[m][k0+khalf+1]
        a.x = s_out[m * LDS_STRIDE + k0 + khalf];
        a.y = s_out[m * LDS_STRIDE + k0 + khalf + 1];
        // B (4x16): lane holds W[k0+khalf][n0+m], W[k0+khalf+1][n0+m]
        bm.x = W[(k0 + khalf)     * D + n0 + m];
        bm.y = W[(k0 + khalf + 1) * D + n0 + m];
        acc = __builtin_amdgcn_wmma_f32_16x16x4_f32(
                /*neg_a=*/false, a, /*neg_b=*/false, bm,
                /*c_mod=*/(short)0, acc, /*reuse_a=*/false, /*reuse_b=*/false);
    }

    // ---- Stage 3: bias + per-graph scatter-add pooling --------------------
    // C/D layout: VGPR r -> M = r + 8*(lane>=16), N = n0 + (lane&15)
    const int   n     = n0 + (lane & 15);
    const float bias  = bvec[n];
    const int   mbase = (lane >> 4) * 8;
    #pragma unroll
    for (int r = 0; r < 8; ++r) {
        int   mrow = mbase + r;
        float val  = acc[r] + bias;
        atomicAdd(pooled2 + (long long)s_batch[mrow] * D + n, val);
    }
}

// ---------------------------------------------------------------------------
extern "C" void kernel_launch(void* const* d_in, const int* in_sizes, int n_in,
                              void* d_out, int out_size, void* d_ws, size_t ws_size,
                              hipStream_t stream) {
    const float* x     = (const float*)d_in[0];
    const float* eps   = (const float*)d_in[1];
    const float* W     = (const float*)d_in[2];
    const float* bvec  = (const float*)d_in[3];
    const int*   ei    = (const int*)d_in[4];
    const int*   batch = (const int*)d_in[5];

    float* out     = (float*)d_out;
    float* pooled2 = out + (size_t)N_NODES * D;      // outputs concatenated flat
    float* agg     = (float*)d_ws;                   // 50000*128 f32 = 25.6 MB

    // 0) zero agg + pooled2
    {
        int total = N_NODES * D + BATCHSIZE * D;
        zero_kernel<<<(total + 255) / 256, 256, 0, stream>>>(agg, pooled2);
    }
    // 1) edge scatter-add (one wave per edge)
    {
        long long threads = (long long)N_EDGES * 32;
        int blocks = (int)((threads + 255) / 256);
        edge_scatter<<<blocks, 256, 0, stream>>>(x, ei, agg);
    }
    // 2) fused relu + WMMA GEMM + pooled scatter
    {
        fused_out_gemm_pool<<<N_NODES / ROWS_PER_BLOCK, 256, 0, stream>>>(
            x, eps, W, bvec, batch, agg, out, pooled2);
    }
}